// GraphSAGEEncoder_79671643341337
// MI455X (gfx1250) — compile-verified
//
#include <hip/hip_runtime.h>
#include <hip/hip_bf16.h>
#include <stdint.h>

#define N_NODES_C  100000
#define N_EDGES_C  1600000
#define N_GRAPHS_C 64
#define IN_C_C     18
#define EDGE_C_C   22
#define HID_C_C    64
#define OUT_C_C    128

typedef float v2f __attribute__((ext_vector_type(2)));
typedef float v8f __attribute__((ext_vector_type(8)));

// ---------------------------------------------------------------------------
// K0: zero all accumulators, seed x1 = x.  Re-run every call (graph replay).
// ---------------------------------------------------------------------------
__global__ void __launch_bounds__(256)
init_kernel(const float* __restrict__ x, float* __restrict__ x1,
            float* __restrict__ deg, float* __restrict__ msum1,
            float* __restrict__ msum2, float* __restrict__ pooled,
            float* __restrict__ cnt) {
    const int i = blockIdx.x * blockDim.x + threadIdx.x;
    if (i < N_NODES_C * HID_C_C) msum2[i] = 0.0f;
    if (i < N_NODES_C * IN_C_C) { x1[i] = x[i]; msum1[i] = 0.0f; }
    if (i < N_NODES_C)          deg[i] = 0.0f;
    if (i < N_GRAPHS_C * HID_C_C) pooled[i] = 0.0f;
    if (i < N_GRAPHS_C)         cnt[i] = 0.0f;
}

// ---------------------------------------------------------------------------
// K1: edge_update = edge_attr @ W + b, scatter-add into x1[src]; degree by dst
// ---------------------------------------------------------------------------
__global__ void __launch_bounds__(256)
edge_fc_scatter(const float* __restrict__ eattr, const int* __restrict__ eindex,
                const float* __restrict__ W, const float* __restrict__ b,
                float* __restrict__ x1, float* __restrict__ deg) {
    __shared__ float sW[EDGE_C_C * IN_C_C];
    __shared__ float sB[IN_C_C];
    for (int i = threadIdx.x; i < EDGE_C_C * IN_C_C; i += blockDim.x) sW[i] = W[i];
    for (int i = threadIdx.x; i < IN_C_C; i += blockDim.x)            sB[i] = b[i];
    __syncthreads();

    const int e = blockIdx.x * blockDim.x + threadIdx.x;
    if (e >= N_EDGES_C) return;
    const int src = eindex[e];
    const int dst = eindex[N_EDGES_C + e];

    float a[EDGE_C_C];
    const float* er = eattr + (size_t)e * EDGE_C_C;
#pragma unroll
    for (int k = 0; k < EDGE_C_C; ++k) a[k] = er[k];

    float* xr = x1 + (size_t)src * IN_C_C;
#pragma unroll
    for (int c = 0; c < IN_C_C; ++c) {
        float acc = sB[c];
#pragma unroll
        for (int k = 0; k < EDGE_C_C; ++k) acc += a[k] * sW[k * IN_C_C + c];
        atomicAdd(xr + c, acc);
    }
    atomicAdd(deg + dst, 1.0f);
}

// ---------------------------------------------------------------------------
// K2/K4: msum[dst] += feat[src]  (neighbor aggregation, L2-resident gathers)
// ---------------------------------------------------------------------------
template <int C>
__global__ void __launch_bounds__(256)
gather_scatter(const int* __restrict__ eindex, const float* __restrict__ feat,
               float* __restrict__ msum) {
    const int e = blockIdx.x * blockDim.x + threadIdx.x;
    if (e >= N_EDGES_C) return;
    const int src = eindex[e];
    const int dst = eindex[N_EDGES_C + e];
    const float* fr = feat + (size_t)src * C;
    float*       md = msum + (size_t)dst * C;
#pragma unroll
    for (int c = 0; c < C; ++c) atomicAdd(md + c, fr[c]);
}

// ---------------------------------------------------------------------------
// K3/K5: SAGEConv dense part with V_WMMA_F32_16X16X4_F32.
// One wave32 per 16-node tile computing all 64 output channels:
//   D = relu( mean(16xK) @ l_w(Kx64) + x(16xK) @ r_w(Kx64) + l_b )
// A frag (16x4 f32): lane L holds A[M=L&15][2*(L>>4) + v], v=0..1
// B frag (4x16 f32): lane L holds B[2*(L>>4) + v][N=L&15]
// C/D frag (16x16):  lane L holds C[v + 8*(L>>4)][N=L&15], v=0..7
// ---------------------------------------------------------------------------
template <int K>
__global__ void __launch_bounds__(64)
sage_wmma_kernel(const float* __restrict__ xin, const float* __restrict__ msum,
                 const float* __restrict__ deg,
                 const float* __restrict__ lw, const float* __restrict__ lb,
                 const float* __restrict__ rw, float* __restrict__ out) {
    const int wave = (blockIdx.x * blockDim.x + threadIdx.x) >> 5;
    const int lane = threadIdx.x & 31;
    if (wave >= N_NODES_C / 16) return;   // exact grid: never taken

    const int node0 = wave * 16;
    const int m    = lane & 15;           // A: M index, B/C/D: N index
    const int half = (lane >> 4) & 1;     // A/B: K-half, C/D: M-half
    const int row  = node0 + m;

    const float dg  = deg[row];
    const float inv = dg > 0.0f ? 1.0f / dg : 0.0f;
    const float* xr = xin  + (size_t)row * K;
    const float* mr = msum + (size_t)row * K;

    v8f acc[4] = {};

    constexpr int KP = (K + 3) & ~3;      // pad K to multiple of 4
#pragma unroll
    for (int ks = 0; ks < KP; ks += 4) {
        const int kb = ks + half * 2;
        v2f amean, aroot;
        amean[0] = (kb     < K) ? mr[kb]     * inv : 0.0f;
        amean[1] = (kb + 1 < K) ? mr[kb + 1] * inv : 0.0f;
        aroot[0] = (kb     < K) ? xr[kb]           : 0.0f;
        aroot[1] = (kb + 1 < K) ? xr[kb + 1]       : 0.0f;
#pragma unroll
        for (int t = 0; t < 4; ++t) {
            const int col = t * 16 + m;
            v2f bl, br;
            bl[0] = (kb     < K) ? lw[(size_t)kb       * HID_C_C + col] : 0.0f;
            bl[1] = (kb + 1 < K) ? lw[(size_t)(kb + 1) * HID_C_C + col] : 0.0f;
            br[0] = (kb     < K) ? rw[(size_t)kb       * HID_C_C + col] : 0.0f;
            br[1] = (kb + 1 < K) ? rw[(size_t)(kb + 1) * HID_C_C + col] : 0.0f;
            acc[t] = __builtin_amdgcn_wmma_f32_16x16x4_f32(
                false, amean, false, bl, (short)0, acc[t], false, false);
            acc[t] = __builtin_amdgcn_wmma_f32_16x16x4_f32(
                false, aroot, false, br, (short)0, acc[t], false, false);
        }
    }

#pragma unroll
    for (int t = 0; t < 4; ++t) {
        const int col  = t * 16 + m;
        const float bias = lb[col];
#pragma unroll
        for (int v = 0; v < 8; ++v) {
            const int orow = node0 + half * 8 + v;
            float val = acc[t][v] + bias;
            out[(size_t)orow * HID_C_C + col] = val > 0.0f ? val : 0.0f;
        }
    }
}

// ---------------------------------------------------------------------------
// K6: global mean pool (sum + count); per-block LDS staging (batch is sorted)
// ---------------------------------------------------------------------------
__global__ void __launch_bounds__(256)
pool_kernel(const float* __restrict__ h2, const int* __restrict__ batch,
            float* __restrict__ pooled, float* __restrict__ cnt) {
    __shared__ float sPool[N_GRAPHS_C * HID_C_C];
    __shared__ float sCnt[N_GRAPHS_C];
    for (int i = threadIdx.x; i < N_GRAPHS_C * HID_C_C; i += blockDim.x) sPool[i] = 0.0f;
    for (int i = threadIdx.x; i < N_GRAPHS_C; i += blockDim.x)           sCnt[i] = 0.0f;
    __syncthreads();

    const int node = blockIdx.x * blockDim.x + threadIdx.x;
    if (node < N_NODES_C) {
        const int g = batch[node];
        atomicAdd(&sCnt[g], 1.0f);
        const float* hr = h2 + (size_t)node * HID_C_C;
#pragma unroll 4
        for (int c = 0; c < HID_C_C; ++c)
            atomicAdd(&sPool[g * HID_C_C + c], hr[c]);
    }
    __syncthreads();

    for (int i = threadIdx.x; i < N_GRAPHS_C * HID_C_C; i += blockDim.x)
        atomicAdd(&pooled[i], sPool[i]);
    for (int i = threadIdx.x; i < N_GRAPHS_C; i += blockDim.x)
        atomicAdd(&cnt[i], sCnt[i]);
}

// ---------------------------------------------------------------------------
// K7: out[g][o] = (pooled[g]/max(cnt,1)) @ proj_w + proj_b    (64x64x128)
// ---------------------------------------------------------------------------
__global__ void __launch_bounds__(256)
proj_kernel(const float* __restrict__ pooled, const float* __restrict__ cnt,
            const float* __restrict__ W, const float* __restrict__ b,
            float* __restrict__ out) {
    const int i = blockIdx.x * blockDim.x + threadIdx.x;
    if (i >= N_GRAPHS_C * OUT_C_C) return;
    const int g = i / OUT_C_C;
    const int o = i % OUT_C_C;
    const float inv = 1.0f / fmaxf(cnt[g], 1.0f);
    float acc = b[o];
    const float* pr = pooled + (size_t)g * HID_C_C;
#pragma unroll 8
    for (int c = 0; c < HID_C_C; ++c)
        acc += (pr[c] * inv) * W[(size_t)c * OUT_C_C + o];
    out[i] = acc;
}

// ---------------------------------------------------------------------------
extern "C" void kernel_launch(void* const* d_in, const int* in_sizes, int n_in,
                              void* d_out, int out_size, void* d_ws, size_t ws_size,
                              hipStream_t stream) {
    const float* x      = (const float*)d_in[0];
    const float* eattr  = (const float*)d_in[1];
    const int*   eindex = (const int*)d_in[2];
    const int*   batch  = (const int*)d_in[3];
    const float* efc_w  = (const float*)d_in[4];
    const float* efc_b  = (const float*)d_in[5];
    const float* c1_lw  = (const float*)d_in[6];
    const float* c1_lb  = (const float*)d_in[7];
    const float* c1_rw  = (const float*)d_in[8];
    const float* c2_lw  = (const float*)d_in[9];
    const float* c2_lb  = (const float*)d_in[10];
    const float* c2_rw  = (const float*)d_in[11];
    const float* proj_w = (const float*)d_in[12];
    const float* proj_b = (const float*)d_in[13];
    float* out = (float*)d_out;

    float* ws = (float*)d_ws;
    size_t off = 0;
    float* x1     = ws + off; off += (size_t)N_NODES_C * IN_C_C;
    float* deg    = ws + off; off += (size_t)N_NODES_C;
    float* msum1  = ws + off; off += (size_t)N_NODES_C * IN_C_C;
    float* h1     = ws + off; off += (size_t)N_NODES_C * HID_C_C;
    float* msum2  = ws + off; off += (size_t)N_NODES_C * HID_C_C;
    float* h2     = ws + off; off += (size_t)N_NODES_C * HID_C_C;
    float* pooled = ws + off; off += (size_t)N_GRAPHS_C * HID_C_C;
    float* cnt    = ws + off; off += (size_t)N_GRAPHS_C;

    init_kernel<<<(N_NODES_C * HID_C_C + 255) / 256, 256, 0, stream>>>(
        x, x1, deg, msum1, msum2, pooled, cnt);

    edge_fc_scatter<<<(N_EDGES_C + 255) / 256, 256, 0, stream>>>(
        eattr, eindex, efc_w, efc_b, x1, deg);

    gather_scatter<IN_C_C><<<(N_EDGES_C + 255) / 256, 256, 0, stream>>>(
        eindex, x1, msum1);

    // 6250 node tiles, 2 waves/block -> 3125 blocks, exact (EXEC all-1s for WMMA)
    sage_wmma_kernel<IN_C_C><<<N_NODES_C / 16 / 2, 64, 0, stream>>>(
        x1, msum1, deg, c1_lw, c1_lb, c1_rw, h1);

    gather_scatter<HID_C_C><<<(N_EDGES_C + 255) / 256, 256, 0, stream>>>(
        eindex, h1, msum2);

    sage_wmma_kernel<HID_C_C><<<N_NODES_C / 16 / 2, 64, 0, stream>>>(
        h1, msum2, deg, c2_lw, c2_lb, c2_rw, h2);

    pool_kernel<<<(N_NODES_C + 255) / 256, 256, 0, stream>>>(
        h2, batch, pooled, cnt);

    proj_kernel<<<(N_GRAPHS_C * OUT_C_C + 255) / 256, 256, 0, stream>>>(
        pooled, cnt, proj_w, proj_b, out);
}